// VectorQuantizer_78469052498032
// MI455X (gfx1250) — compile-verified
//
#include <hip/hip_runtime.h>
#include <math.h>

typedef float v2f __attribute__((ext_vector_type(2)));
typedef float v8f __attribute__((ext_vector_type(8)));
typedef int   v2i_ll __attribute__((vector_size(8)));   // matches async-LDS builtin param

#define N_TOK 65536
#define N_E   1024
#define E_DIM 256
#define BETA  0.25f

// GEMM tiling
#define BM 128
#define BN 128
#define BK 16
#define KP (BK/2)   // float2 k-pair rows per chunk
#define NCHUNK (E_DIM / BK)

// d_out float offsets (return order: loss, z_q_st, perplexity, d, emb_w)
#define OUT_LOSS 0
#define OUT_ZQ   1
#define OUT_PERP (1 + N_TOK*E_DIM)               // 16777217
#define OUT_D    (2 + N_TOK*E_DIM)               // 16777218
#define OUT_EMB  (2 + N_TOK*E_DIM + N_TOK*N_E)   // 83886082

#define AS1 __attribute__((address_space(1)))
#define AS3 __attribute__((address_space(3)))

#if defined(__has_builtin)
#if __has_builtin(__builtin_amdgcn_global_load_async_to_lds_b64)
#define HAVE_ASYNC_LDS 1
#endif
#endif

__device__ __forceinline__ void wait_async_all() {
#ifdef HAVE_ASYNC_LDS
#if __has_builtin(__builtin_amdgcn_s_wait_asynccnt)
    __builtin_amdgcn_s_wait_asynccnt(0);
#else
    asm volatile("s_wait_asynccnt 0x0" ::: "memory");
#endif
#endif
}

__global__ void vq_init(unsigned* __restrict__ counts) {
    int t = blockIdx.x * blockDim.x + threadIdx.x;
    if (t < N_E) counts[t] = 0u;
}

// One wave per row: sum of squares of a 256-float row.
__global__ void vq_rownorm(const float* __restrict__ x, float* __restrict__ out, int nrows) {
    int gid  = blockIdx.x * blockDim.x + threadIdx.x;
    int wid  = gid >> 5;
    int lane = gid & 31;
    if (wid >= nrows) return;
    const float4* row = (const float4*)(x + (size_t)wid * E_DIM);
    float s = 0.f;
#pragma unroll
    for (int i = 0; i < 2; ++i) {
        float4 v = row[lane + i * 32];
        s += v.x * v.x + v.y * v.y + v.z * v.z + v.w * v.w;
    }
#pragma unroll
    for (int off = 16; off > 0; off >>= 1) s += __shfl_xor(s, off, 32);
    if (lane == 0) out[wid] = s;
}

// d = z2[m] + e2[n] - 2 * z @ emb^T  via V_WMMA_F32_16X16X4_F32, plus fused
// per-(row, col-block) argmin partials so the 268MB d matrix is never re-read.
// Block: 256 threads = 8 waves, tile 128(M) x 128(N), K double-buffered by 16
// through LDS using async global->LDS DMA (ASYNCcnt) when available.
// Waves arranged 4(M) x 2(N): each wave owns 32 rows x 64 cols = 2x4 16x16 tiles.
__global__ __launch_bounds__(256) void vq_dist_wmma(
    const float* __restrict__ z, const float* __restrict__ emb,
    const float* __restrict__ z2, const float* __restrict__ e2,
    float* __restrict__ dmat,
    float* __restrict__ argv, int* __restrict__ argn)
{
    // pair-interleaved transposed tiles: lds_a[b][kp][m] = {z[m][2kp], z[m][2kp+1]}
    __shared__ float2 lds_a[2][KP][BM + 1];
    __shared__ float2 lds_b[2][KP][BN + 1];
    __shared__ float  lz2[BM];
    __shared__ float  le2[BN];
    __shared__ float  redv[BM][2];
    __shared__ int    redn[BM][2];

    const int tid  = threadIdx.x;
    const int lane = tid & 31;
    const int wid  = tid >> 5;
    const int wm   = wid & 3;    // M group (x32 rows)
    const int wn   = wid >> 2;   // N group (x64 cols)
    const int row_base = blockIdx.y * BM;
    const int col_base = blockIdx.x * BN;
    const int half = lane >> 4;
    const int l16  = lane & 31 & 15;

    if (tid < BM)           lz2[tid]      = z2[row_base + tid];
    else if (tid < BM + BN) le2[tid - BM] = e2[col_base + (tid - BM)];

    v8f acc[2][4];
#pragma unroll
    for (int i = 0; i < 2; ++i)
#pragma unroll
        for (int j = 0; j < 4; ++j) { v8f zero = {}; acc[i][j] = zero; }

    // Stage one BK-chunk of A (z) and B (emb), transposed into pair layout.
    auto issue = [&](int buf, int kb) {
#pragma unroll
        for (int i = 0; i < 4; ++i) {
            int lin = tid + i * 256;        // 0..1023
            int kp  = lin & (KP - 1);       // pair row 0..7
            int r   = lin >> 3;             // tile row 0..127
            const float* ga = z   + (size_t)(row_base + r) * E_DIM + kb + 2 * kp;
            const float* gb = emb + (size_t)(col_base + r) * E_DIM + kb + 2 * kp;
#ifdef HAVE_ASYNC_LDS
            __builtin_amdgcn_global_load_async_to_lds_b64(
                (AS1 v2i_ll*)ga, (AS3 v2i_ll*)&lds_a[buf][kp][r], 0, 0);
            __builtin_amdgcn_global_load_async_to_lds_b64(
                (AS1 v2i_ll*)gb, (AS3 v2i_ll*)&lds_b[buf][kp][r], 0, 0);
#else
            lds_a[buf][kp][r] = *(const float2*)ga;
            lds_b[buf][kp][r] = *(const float2*)gb;
#endif
        }
    };

    issue(0, 0);
    for (int kc = 0; kc < NCHUNK; ++kc) {
        const int cur = kc & 1;
        wait_async_all();      // my portion of chunk kc landed in LDS
        __syncthreads();       // everyone's portion landed; prev compute done
        if (kc + 1 < NCHUNK) issue(cur ^ 1, (kc + 1) * BK);  // overlaps WMMAs below
#pragma unroll
        for (int ks = 0; ks < BK / 4; ++ks) {
            const int p = 2 * ks + half;   // lanes 0-15: k0,k0+1 ; lanes 16-31: k0+2,k0+3
            v2f a0 = *(const v2f*)&lds_a[cur][p][wm * 32 +      l16];
            v2f a1 = *(const v2f*)&lds_a[cur][p][wm * 32 + 16 + l16];
            v2f b0 = *(const v2f*)&lds_b[cur][p][wn * 64 +      l16];
            v2f b1 = *(const v2f*)&lds_b[cur][p][wn * 64 + 16 + l16];
            v2f b2 = *(const v2f*)&lds_b[cur][p][wn * 64 + 32 + l16];
            v2f b3 = *(const v2f*)&lds_b[cur][p][wn * 64 + 48 + l16];
            acc[0][0] = __builtin_amdgcn_wmma_f32_16x16x4_f32(false, a0, false, b0, (short)0, acc[0][0], false, false);
            acc[0][1] = __builtin_amdgcn_wmma_f32_16x16x4_f32(false, a0, false, b1, (short)0, acc[0][1], false, false);
            acc[0][2] = __builtin_amdgcn_wmma_f32_16x16x4_f32(false, a0, false, b2, (short)0, acc[0][2], false, false);
            acc[0][3] = __builtin_amdgcn_wmma_f32_16x16x4_f32(false, a0, false, b3, (short)0, acc[0][3], false, false);
            acc[1][0] = __builtin_amdgcn_wmma_f32_16x16x4_f32(false, a1, false, b0, (short)0, acc[1][0], false, false);
            acc[1][1] = __builtin_amdgcn_wmma_f32_16x16x4_f32(false, a1, false, b1, (short)0, acc[1][1], false, false);
            acc[1][2] = __builtin_amdgcn_wmma_f32_16x16x4_f32(false, a1, false, b2, (short)0, acc[1][2], false, false);
            acc[1][3] = __builtin_amdgcn_wmma_f32_16x16x4_f32(false, a1, false, b3, (short)0, acc[1][3], false, false);
        }
    }

    // Epilogue: d = z2 + e2 - 2*acc; track per-(i,r) argmin over this block's cols.
    // C/D layout: VGPR r, lanes<16 -> M=r, lanes>=16 -> M=r+8; N = lane&15.
    float bv[2][8];
    int   bn[2][8];
#pragma unroll
    for (int i = 0; i < 2; ++i)
#pragma unroll
        for (int r = 0; r < 8; ++r) { bv[i][r] = 3.402823466e38f; bn[i][r] = 0; }

#pragma unroll
    for (int i = 0; i < 2; ++i)
#pragma unroll
        for (int j = 0; j < 4; ++j)   // ascending j => ascending n => first-occurrence ties
#pragma unroll
            for (int r = 0; r < 8; ++r) {
                int ml = wm * 32 + i * 16 + r + 8 * half;
                int nl = wn * 64 + j * 16 + l16;
                float v = lz2[ml] + le2[nl] - 2.0f * acc[i][j][r];
                dmat[(size_t)(row_base + ml) * N_E + col_base + nl] = v;
                if (v < bv[i][r]) { bv[i][r] = v; bn[i][r] = col_base + nl; }
            }

    // Cross-lane min within each 16-lane half (same row set), tie-break lower n.
#pragma unroll
    for (int i = 0; i < 2; ++i)
#pragma unroll
        for (int r = 0; r < 8; ++r) {
            float b = bv[i][r];
            int   n = bn[i][r];
#pragma unroll
            for (int off = 1; off < 16; off <<= 1) {
                float ob = __shfl_xor(b, off, 32);
                int   on = __shfl_xor(n, off, 32);
                if (ob < b || (ob == b && on < n)) { b = ob; n = on; }
            }
            if (l16 == 0) {
                int ml = wm * 32 + i * 16 + r + 8 * half;
                redv[ml][wn] = b;
                redn[ml][wn] = n;
            }
        }
    __syncthreads();
    if (tid < BM) {
        float b0 = redv[tid][0]; int n0 = redn[tid][0];
        float b1 = redv[tid][1]; int n1 = redn[tid][1];
        if (b1 < b0 || (b1 == b0 && n1 < n0)) { b0 = b1; n0 = n1; }
        argv[(size_t)(row_base + tid) * 8 + blockIdx.x] = b0;
        argn[(size_t)(row_base + tid) * 8 + blockIdx.x] = n0;
    }
}

// Reduce the 8 per-column-block partials per row (ascending block => tie-break ok).
__global__ void vq_argmin_final(const float* __restrict__ argv, const int* __restrict__ argn,
                                int* __restrict__ idx) {
    int m = blockIdx.x * blockDim.x + threadIdx.x;
    if (m >= N_TOK) return;
    const float* av = argv + (size_t)m * 8;
    const int*   an = argn + (size_t)m * 8;
    float best = av[0]; int bi = an[0];
#pragma unroll
    for (int j = 1; j < 8; ++j) {
        float v = av[j]; int n = an[j];
        if (v < best || (v == best && n < bi)) { best = v; bi = n; }
    }
    idx[m] = bi;
}

// One wave per row: z_q_st = emb[idx[m]]; deterministic per-block SSE partial; int histogram.
__global__ void vq_gather(const float* __restrict__ z, const float* __restrict__ emb,
                          const int* __restrict__ idx, float* __restrict__ zq_st,
                          unsigned* __restrict__ counts, float* __restrict__ partials)
{
    __shared__ float wsum[8];
    int tid  = threadIdx.x;
    int lane = tid & 31;
    int w    = tid >> 5;
    int m    = blockIdx.x * 8 + w;
    int code = idx[m];
    const float* e  = emb + (size_t)code * E_DIM;
    const float* zr = z   + (size_t)m * E_DIM;
    float*       o  = zq_st + (size_t)m * E_DIM;
    float s = 0.f;
#pragma unroll
    for (int i = 0; i < 8; ++i) {
        int c = lane + i * 32;
        float ev = e[c];
        float zv = zr[c];
        o[c] = ev;
        float df = ev - zv;
        s += df * df;
    }
#pragma unroll
    for (int off = 16; off > 0; off >>= 1) s += __shfl_xor(s, off, 32);
    if (lane == 0) { wsum[w] = s; atomicAdd(&counts[code], 1u); }
    __syncthreads();
    if (tid == 0) {
        float t = 0.f;
#pragma unroll
        for (int i = 0; i < 8; ++i) t += wsum[i];
        partials[blockIdx.x] = t;
    }
}

// Single block: fixed-order reductions (deterministic) for loss + perplexity.
__global__ void vq_final(const float* __restrict__ partials, const unsigned* __restrict__ counts,
                         float* __restrict__ out_loss, float* __restrict__ out_perp)
{
    __shared__ float red[256];
    int tid = threadIdx.x;
    float s = 0.f;
    for (int i = tid; i < N_TOK / 8; i += 256) s += partials[i];
    red[tid] = s;
    __syncthreads();
    for (int st = 128; st > 0; st >>= 1) { if (tid < st) red[tid] += red[tid + st]; __syncthreads(); }
    float sse = red[0];
    __syncthreads();
    float ent = 0.f;
    for (int i = tid; i < N_E; i += 256) {
        float em = (float)counts[i] * (1.0f / (float)N_TOK);
        ent += em * logf(em + 1e-10f);
    }
    red[tid] = ent;
    __syncthreads();
    for (int st = 128; st > 0; st >>= 1) { if (tid < st) red[tid] += red[tid + st]; __syncthreads(); }
    if (tid == 0) {
        *out_loss = (1.0f + BETA) * sse / (float)(N_TOK * E_DIM);
        *out_perp = expf(-red[0]);
    }
}

extern "C" void kernel_launch(void* const* d_in, const int* in_sizes, int n_in,
                              void* d_out, int out_size, void* d_ws, size_t ws_size,
                              hipStream_t stream) {
    const float* z   = (const float*)d_in[0];
    const float* emb = (const float*)d_in[1];
    float* out = (float*)d_out;

    // ws layout (floats): z2[65536] | e2[1024] | idx[65536] | counts[1024] |
    //                     partials[8192] | argv[65536*8] | argn[65536*8]
    float*    w        = (float*)d_ws;
    float*    z2       = w;
    float*    e2       = w + N_TOK;
    int*      idx      = (int*)(w + N_TOK + N_E);
    unsigned* counts   = (unsigned*)(w + 2 * N_TOK + N_E);
    float*    partials = w + 2 * N_TOK + 2 * N_E;
    float*    argv     = partials + N_TOK / 8;
    int*      argn     = (int*)(argv + (size_t)N_TOK * 8);

    float* loss_p = out + OUT_LOSS;
    float* zq_st  = out + OUT_ZQ;
    float* perp_p = out + OUT_PERP;
    float* dmat   = out + OUT_D;
    float* embout = out + OUT_EMB;

    vq_init<<<4, 256, 0, stream>>>(counts);
    vq_rownorm<<<(N_TOK * 32) / 256, 256, 0, stream>>>(z, z2, N_TOK);
    vq_rownorm<<<(N_E * 32) / 256, 256, 0, stream>>>(emb, e2, N_E);

    dim3 grid(N_E / BN, N_TOK / BM);   // (8, 512)
    vq_dist_wmma<<<grid, 256, 0, stream>>>(z, emb, z2, e2, dmat, argv, argn);

    vq_argmin_final<<<N_TOK / 256, 256, 0, stream>>>(argv, argn, idx);
    vq_gather<<<N_TOK / 8, 256, 0, stream>>>(z, emb, idx, zq_st, counts, partials);
    vq_final<<<1, 256, 0, stream>>>(partials, counts, loss_p, perp_p);

    (void)hipMemcpyAsync(embout, (const void*)emb, (size_t)N_E * E_DIM * sizeof(float),
                         hipMemcpyDeviceToDevice, stream);
}